// GNNWithGlobalFeats_52510270160896
// MI455X (gfx1250) — compile-verified
//
#include <hip/hip_runtime.h>

typedef __attribute__((ext_vector_type(2))) float v2f;
typedef __attribute__((ext_vector_type(8))) float v8f;

#define NNODES 50000
#define NEDGES 800000
#define NBATCH 512
#define HDIM   128
#define GDIM   32
#define EPS_BN 1e-5f

// ---------------------------------------------------------------------------
// WMMA GEMM: out[rows, NT*16] = A[rows, K] @ W[K, NT*16] (+bias) (opt ReLU)
// rows must be a multiple of 16. One wave computes a 16 x (NT*16) strip using
// V_WMMA_F32_16X16X4_F32; W is staged through LDS in 64-row chunks.
// Fragment layouts per CDNA5 ISA 7.12.2:
//   A 16x4 f32 : lane L holds A[L%16][2*(L/16)+0 .. +1]  (v0,v1)
//   B 4x16 f32 : lane L holds B[2*(L/16)+0][L%16], B[2*(L/16)+1][L%16]
//   C/D 16x16  : VGPR r, lane L -> row (r + 8*(L/16)), col L%16
// ---------------------------------------------------------------------------
template <int K, int NT>
__global__ __launch_bounds__(256) void gemm_wmma_f32(
    const float* __restrict__ A, const float* __restrict__ W,
    const float* __restrict__ bias, float* __restrict__ out,
    int rows, int relu) {
  constexpr int NC  = NT * 16;
  constexpr int KC  = 64;        // K-chunk staged in LDS
  constexpr int LDW = NC + 4;    // pad: (NC+4) % 64 == 4 -> no bank conflicts
  __shared__ float wlds[KC * LDW];

  const int wave = threadIdx.x >> 5;
  const int lane = threadIdx.x & 31;
  const int lrow = lane & 15;    // row-in-tile (A) / col-in-tile (B,C)
  const int half = lane >> 4;    // K half selector

  const int row0   = (blockIdx.x * 8 + wave) * 16;
  const bool active = (row0 < rows);

  v8f acc[NT];
#pragma unroll
  for (int t = 0; t < NT; ++t) acc[t] = (v8f){0.f,0.f,0.f,0.f,0.f,0.f,0.f,0.f};

  const float* arow = A + (size_t)(row0 + lrow) * K;

  for (int kc = 0; kc < K; kc += KC) {
    const int krem = (K - kc < KC) ? (K - kc) : KC;
    __syncthreads();
    for (int i = threadIdx.x; i < krem * NC; i += 256) {
      int r = i / NC, c = i - r * NC;
      wlds[r * LDW + c] = W[(size_t)(kc + r) * NC + c];
    }
    __syncthreads();

    if (active) {
      for (int k = 0; k < krem; k += 4) {
        const float2 av = *(const float2*)(arow + kc + k + 2 * half);
        v2f a; a.x = av.x; a.y = av.y;
#pragma unroll
        for (int t = 0; t < NT; ++t) {
          v2f b;
          b.x = wlds[(k + 2 * half + 0) * LDW + t * 16 + lrow];
          b.y = wlds[(k + 2 * half + 1) * LDW + t * 16 + lrow];
          acc[t] = __builtin_amdgcn_wmma_f32_16x16x4_f32(
              false, a, false, b, (short)0, acc[t], false, false);
        }
      }
    }
  }

  if (!active) return;
#pragma unroll
  for (int t = 0; t < NT; ++t) {
    const int col = t * 16 + lrow;
    const float bv = bias ? bias[col] : 0.f;
#pragma unroll
    for (int r = 0; r < 8; ++r) {
      const int row = row0 + half * 8 + r;
      float v = acc[t][r] + bv;
      if (relu) v = fmaxf(v, 0.f);
      out[(size_t)row * NC + col] = v;
    }
  }
}

// ---------------------------------------------------------------------------
// Graph helpers
// ---------------------------------------------------------------------------
__global__ void k_fill(float* __restrict__ p, float v, int n) {
  int i = blockIdx.x * blockDim.x + threadIdx.x;
  if (i < n) p[i] = v;
}

__global__ void k_deg_count(const int* __restrict__ dst, float* __restrict__ deg, int e) {
  int i = blockIdx.x * blockDim.x + threadIdx.x;
  if (i < e) atomicAdd(&deg[dst[i]], 1.0f);
}

__global__ void k_deg_rsqrt(float* __restrict__ d, int n) {
  int i = blockIdx.x * blockDim.x + threadIdx.x;
  if (i < n) d[i] = rsqrtf(fmaxf(d[i], 1.0f));
}

// out[i,c] = bias[c] + dinv[i]^2 * hW[i,c]   (self-loop term + GCN bias)
__global__ void k_selfloop(const float* __restrict__ hW, const float* __restrict__ dinv,
                           const float* __restrict__ bias, float* __restrict__ out, int total) {
  int i = blockIdx.x * blockDim.x + threadIdx.x;
  if (i >= total) return;
  int node = i >> 7, c = i & 127;
  float di = dinv[node];
  out[i] = bias[c] + di * di * hW[i];
}

// one wave per edge; each lane moves 4 channels (float4 = 512B/wave gather)
__global__ __launch_bounds__(256) void k_aggregate(
    const float* __restrict__ hW, const int* __restrict__ src,
    const int* __restrict__ dst, const float* __restrict__ dinv,
    float* __restrict__ out, int E) {
  int gid = blockIdx.x * 256 + threadIdx.x;
  int e = gid >> 5, lane = gid & 31;
  if (e >= E) return;
  int s = src[e], d = dst[e];
  float w = dinv[s] * dinv[d];
  float4 v = ((const float4*)(hW + (size_t)s * HDIM))[lane];
  float* ob = out + (size_t)d * HDIM + lane * 4;
  atomicAdd(ob + 0, w * v.x);
  atomicAdd(ob + 1, w * v.y);
  atomicAdd(ob + 2, w * v.z);
  atomicAdd(ob + 3, w * v.w);
}

// one block per channel: mean + inv-std over N rows
__global__ __launch_bounds__(256) void k_bnstats(
    const float* __restrict__ h, float* __restrict__ mean,
    float* __restrict__ istd, int n) {
  __shared__ float s1[256], s2[256];
  int c = blockIdx.x;
  float sum = 0.f, sq = 0.f;
  for (int i = threadIdx.x; i < n; i += 256) {
    float v = h[(size_t)i * HDIM + c];
    sum += v; sq += v * v;
  }
  s1[threadIdx.x] = sum; s2[threadIdx.x] = sq;
  __syncthreads();
  for (int off = 128; off > 0; off >>= 1) {
    if (threadIdx.x < off) {
      s1[threadIdx.x] += s1[threadIdx.x + off];
      s2[threadIdx.x] += s2[threadIdx.x + off];
    }
    __syncthreads();
  }
  if (threadIdx.x == 0) {
    float mu  = s1[0] / (float)n;
    float var = s2[0] / (float)n - mu * mu;
    mean[c] = mu;
    istd[c] = rsqrtf(var + EPS_BN);
  }
}

__global__ void k_bnrelu(const float* __restrict__ h, const float* __restrict__ mean,
                         const float* __restrict__ istd, const float* __restrict__ gamma,
                         const float* __restrict__ beta, float* __restrict__ out, int total) {
  int i = blockIdx.x * blockDim.x + threadIdx.x;
  if (i >= total) return;
  int c = i & 127;
  float v = (h[i] - mean[c]) * istd[c] * gamma[c] + beta[c];
  out[i] = fmaxf(v, 0.f);
}

// one wave per node: segment-sum into [B,H] + counts
__global__ __launch_bounds__(256) void k_pool(
    const float* __restrict__ h, const int* __restrict__ batch,
    float* __restrict__ sums, float* __restrict__ cnt, int n) {
  int gid = blockIdx.x * 256 + threadIdx.x;
  int node = gid >> 5, lane = gid & 31;
  if (node >= n) return;
  int b = batch[node];
  float4 v = ((const float4*)(h + (size_t)node * HDIM))[lane];
  float* ob = sums + (size_t)b * HDIM + lane * 4;
  atomicAdd(ob + 0, v.x);
  atomicAdd(ob + 1, v.y);
  atomicAdd(ob + 2, v.z);
  atomicAdd(ob + 3, v.w);
  if (lane == 0) atomicAdd(&cnt[b], 1.0f);
}

// z = [pooled | global_feats] : [B, 160]
__global__ void k_concat(const float* __restrict__ sums, const float* __restrict__ cnt,
                         const float* __restrict__ gf, float* __restrict__ z) {
  int i = blockIdx.x * blockDim.x + threadIdx.x;
  if (i >= NBATCH * (HDIM + GDIM)) return;
  int b = i / (HDIM + GDIM), c = i - b * (HDIM + GDIM);
  z[i] = (c < HDIM) ? sums[b * HDIM + c] / fmaxf(cnt[b], 1.f)
                    : gf[b * GDIM + (c - HDIM)];
}

// out[b] = z2[b,:] . w + bias
__global__ void k_final(const float* __restrict__ z2, const float* __restrict__ w,
                        const float* __restrict__ b, float* __restrict__ out) {
  int i = blockIdx.x * blockDim.x + threadIdx.x;
  if (i >= NBATCH) return;
  float acc = b[0];
#pragma unroll
  for (int k = 0; k < 64; ++k) acc += z2[i * 64 + k] * w[k];
  out[i] = acc;
}

// ---------------------------------------------------------------------------
extern "C" void kernel_launch(void* const* d_in, const int* in_sizes, int n_in,
                              void* d_out, int out_size, void* d_ws, size_t ws_size,
                              hipStream_t stream) {
  (void)in_sizes; (void)n_in; (void)out_size; (void)ws_size;
  const float* x     = (const float*)d_in[0];
  const int*   eidx  = (const int*)d_in[1];
  const int*   batch = (const int*)d_in[2];
  const float* gf    = (const float*)d_in[3];
  const float* Wl[3]  = {(const float*)d_in[4],  (const float*)d_in[8],  (const float*)d_in[12]};
  const float* bl[3]  = {(const float*)d_in[5],  (const float*)d_in[9],  (const float*)d_in[13]};
  const float* gl[3]  = {(const float*)d_in[6],  (const float*)d_in[10], (const float*)d_in[14]};
  const float* bel[3] = {(const float*)d_in[7],  (const float*)d_in[11], (const float*)d_in[15]};
  const float* M1w = (const float*)d_in[16]; const float* M1b = (const float*)d_in[17];
  const float* M2w = (const float*)d_in[18]; const float* M2b = (const float*)d_in[19];
  const float* M3w = (const float*)d_in[20]; const float* M3b = (const float*)d_in[21];
  const int* src = eidx;
  const int* dst = eidx + NEDGES;

  // workspace layout (floats)
  float* ws   = (float*)d_ws;
  float* bufA = ws;                        // [N,128] activations
  float* bufB = bufA + (size_t)NNODES * HDIM;   // [N,128] h@W
  float* bufC = bufB + (size_t)NNODES * HDIM;   // [N,128] aggregated
  float* dinv = bufC + (size_t)NNODES * HDIM;   // [N]
  float* mean = dinv + NNODES;             // [128]
  float* istd = mean + HDIM;               // [128]
  float* sums = istd + HDIM;               // [B,128]
  float* cnt  = sums + (size_t)NBATCH * HDIM;   // [B]
  float* z    = cnt + NBATCH;              // [B,160]
  float* z1   = z + (size_t)NBATCH * (HDIM + GDIM); // [B,128]
  float* z2   = z1 + (size_t)NBATCH * HDIM;         // [B,64]

  const int T = 256;
  // degree -> dinv = rsqrt(in_deg + 1)
  k_fill<<<(NNODES + T - 1) / T, T, 0, stream>>>(dinv, 1.0f, NNODES);
  k_deg_count<<<(NEDGES + T - 1) / T, T, 0, stream>>>(dst, dinv, NEDGES);
  k_deg_rsqrt<<<(NNODES + T - 1) / T, T, 0, stream>>>(dinv, NNODES);

  const int gemm_blocks_N = ((NNODES / 16) + 7) / 8;   // 3125 strips -> 391 blocks
  const int total = NNODES * HDIM;
  const float* hin = x;
  for (int l = 0; l < 3; ++l) {
    gemm_wmma_f32<128, 8><<<gemm_blocks_N, T, 0, stream>>>(hin, Wl[l], nullptr, bufB, NNODES, 0);
    k_selfloop<<<(total + T - 1) / T, T, 0, stream>>>(bufB, dinv, bl[l], bufC, total);
    k_aggregate<<<(NEDGES * 32 + T - 1) / T, T, 0, stream>>>(bufB, src, dst, dinv, bufC, NEDGES);
    k_bnstats<<<HDIM, T, 0, stream>>>(bufC, mean, istd, NNODES);
    k_bnrelu<<<(total + T - 1) / T, T, 0, stream>>>(bufC, mean, istd, gl[l], bel[l], bufA, total);
    hin = bufA;
  }

  // mean pool + concat
  hipMemsetAsync(sums, 0, (size_t)(NBATCH * HDIM + NBATCH) * sizeof(float), stream);
  k_pool<<<(NNODES * 32 + T - 1) / T, T, 0, stream>>>(bufA, batch, sums, cnt, NNODES);
  k_concat<<<(NBATCH * (HDIM + GDIM) + T - 1) / T, T, 0, stream>>>(sums, cnt, gf, z);

  // MLP head
  const int gemm_blocks_B = ((NBATCH / 16) + 7) / 8;   // 32 strips -> 4 blocks
  gemm_wmma_f32<160, 8><<<gemm_blocks_B, T, 0, stream>>>(z,  M1w, M1b, z1, NBATCH, 1);
  gemm_wmma_f32<128, 4><<<gemm_blocks_B, T, 0, stream>>>(z1, M2w, M2b, z2, NBATCH, 1);
  k_final<<<(NBATCH + T - 1) / T, T, 0, stream>>>(z2, M3w, M3b, (float*)d_out);
}